// TI_TransformerNoEdges_39479339384971
// MI455X (gfx1250) — compile-verified
//
#include <hip/hip_runtime.h>
#include <math.h>

// ---------------------------------------------------------------------------
// Problem constants (match reference)
// ---------------------------------------------------------------------------
#define NN      16384      // total nodes
#define NGRAPH  16
#define NODES   1024
#define DM      256
#define HH      4          // GAT heads
#define CC      64         // GAT head dim
#define NHEAD   2
#define DH      128
#define FF      1024
#define OUTD    1024

typedef __attribute__((ext_vector_type(2))) float v2f;
typedef __attribute__((ext_vector_type(8))) float v8f;

#define WMMA_F32(a, b, c) \
  __builtin_amdgcn_wmma_f32_16x16x4_f32(false, (a), false, (b), (short)0, (c), false, false)

// ---------------------------------------------------------------------------
// Generic fp32 WMMA GEMM:  C[M,N] = A[M,K] @ B[K,N] (+bias) (+act)
// One wave computes a 16(M) x 64(N) tile.  grid = (M/16, N/64), block = 32.
// act: 0 = none, 1 = leaky(0.01), 2 = relu
// ---------------------------------------------------------------------------
__global__ void gemm_wmma_f32(const float* __restrict__ A,
                              const float* __restrict__ Bw,
                              const float* __restrict__ bias,
                              float* __restrict__ Co,
                              int M, int N, int K, int act) {
  const int lane = threadIdx.x & 31;
  const int lo = lane & 15, hi = lane >> 4;
  const int m0 = blockIdx.x * 16;
  const int n0 = blockIdx.y * 64;
  const int row = m0 + lo;

  v8f acc[4] = {};
  for (int k = 0; k < K; k += 4) {
    v2f a;
    const float* ap = A + (size_t)row * K + k + 2 * hi;
    a.x = ap[0];
    a.y = ap[1];
#pragma unroll
    for (int t = 0; t < 4; ++t) {
      const int col = n0 + t * 16 + lo;
      v2f b;
      b.x = Bw[(size_t)(k + 2 * hi) * N + col];
      b.y = Bw[(size_t)(k + 2 * hi + 1) * N + col];
      acc[t] = WMMA_F32(a, b, acc[t]);
    }
  }
#pragma unroll
  for (int t = 0; t < 4; ++t) {
    const int col = n0 + t * 16 + lo;
    const float bv = bias ? bias[col] : 0.0f;
#pragma unroll
    for (int r = 0; r < 8; ++r) {
      const int m = m0 + r + 8 * hi;
      float v = acc[t][r] + bv;
      if (act == 1)      v = v > 0.0f ? v : 0.01f * v;
      else if (act == 2) v = v > 0.0f ? v : 0.0f;
      Co[(size_t)m * N + col] = v;
    }
  }
}

// ---------------------------------------------------------------------------
// GAT: per-node attention scores  s_src[n,h] = sum_c H[n,h*64+c]*a_s[h,c]
// ---------------------------------------------------------------------------
__global__ void gat_scores(const float* __restrict__ H,
                           const float* __restrict__ a_s,
                           const float* __restrict__ a_d,
                           float* __restrict__ ss, float* __restrict__ sd) {
  const int idx = blockIdx.x * blockDim.x + threadIdx.x;  // n*HH + h
  if (idx >= NN * HH) return;
  const int n = idx >> 2, h = idx & 3;
  float s1 = 0.0f, s2 = 0.0f;
  const float* hp = H + (size_t)n * DM + h * CC;
  const float* asp = a_s + h * CC;
  const float* adp = a_d + h * CC;
  for (int c = 0; c < CC; ++c) {
    const float hv = hp[c];
    s1 += hv * asp[c];
    s2 += hv * adp[c];
  }
  ss[idx] = s1;
  sd[idx] = s2;
}

// ordered-uint encoding so unsigned atomicMax == float max
__device__ __forceinline__ unsigned fkey(float f) {
  unsigned b = __float_as_uint(f);
  return b ^ (unsigned)(((int)b >> 31) | 0x80000000);
}
__device__ __forceinline__ float fdec(unsigned k) {
  unsigned b = k ^ ((k >> 31) ? 0x80000000u : 0xFFFFFFFFu);
  return __uint_as_float(b);
}

__global__ void gat_edge1(const int* __restrict__ src, const int* __restrict__ dst,
                          const float* __restrict__ ss, const float* __restrict__ sd,
                          float* __restrict__ eb, unsigned* __restrict__ mkey, int E) {
  const int idx = blockIdx.x * blockDim.x + threadIdx.x;  // e*HH + h
  if (idx >= E * HH) return;
  const int e = idx >> 2, h = idx & 3;
  const int s = src[e], d = dst[e];
  float v = ss[s * HH + h] + sd[d * HH + h];
  v = v > 0.0f ? v : 0.2f * v;          // leaky_relu(., 0.2)
  eb[idx] = v;
  atomicMax(&mkey[d * HH + h], fkey(v));
}

__global__ void gat_edge2(const int* __restrict__ dst,
                          float* __restrict__ eb, const unsigned* __restrict__ mkey,
                          float* __restrict__ denom, int E) {
  const int idx = blockIdx.x * blockDim.x + threadIdx.x;
  if (idx >= E * HH) return;
  const int e = idx >> 2, h = idx & 3;
  const int d = dst[e];
  const float m = fdec(mkey[d * HH + h]);
  const float p = __expf(eb[idx] - m);
  eb[idx] = p;
  atomicAdd(&denom[d * HH + h], p);
}

__global__ void gat_edge3(const int* __restrict__ src, const int* __restrict__ dst,
                          const float* __restrict__ eb, const float* __restrict__ denom,
                          const float* __restrict__ H, float* __restrict__ Out, int E) {
  const long long idx = (long long)blockIdx.x * blockDim.x + threadIdx.x;  // e*256 + c
  if (idx >= (long long)E * DM) return;
  const int e = (int)(idx >> 8);
  const int c = (int)(idx & 255);
  const int h = c >> 6;
  const int s = src[e], d = dst[e];
  const float alpha = eb[e * HH + h] / (denom[d * HH + h] + 1e-16f);
  atomicAdd(&Out[(size_t)d * DM + c], alpha * H[(size_t)s * DM + c]);
}

__global__ void bias_leaky(float* __restrict__ X, const float* __restrict__ b, int total) {
  const int i = blockIdx.x * blockDim.x + threadIdx.x;
  if (i >= total) return;
  const float v = X[i] + b[i & (DM - 1)];
  X[i] = v > 0.0f ? v : 0.01f * v;      // gat bias then leaky(0.01)
}

// ---------------------------------------------------------------------------
// Flash attention: one wave per (graph, head, 16-query tile).
// Q,K,V,O are [16384, 256] with head h at columns [h*128, (h+1)*128).
// ---------------------------------------------------------------------------
__global__ void flash_attn(const float* __restrict__ Q, const float* __restrict__ Km,
                           const float* __restrict__ Vm, float* __restrict__ O) {
  const int lane = threadIdx.x & 31;
  const int lo = lane & 15, hi = lane >> 4;
  const int g = blockIdx.z, h = blockIdx.y;
  const int q0 = blockIdx.x * 16;
  const int rowbase = g * NODES + q0;
  const int cbase = h * DH;
  __shared__ float Pt[16][17];

  // Q tile as 32 A-fragments (K-dim = 128)
  v2f qf[32];
#pragma unroll
  for (int kc = 0; kc < 32; ++kc) {
    const float* p = Q + (size_t)(rowbase + lo) * DM + cbase + kc * 4 + 2 * hi;
    qf[kc].x = p[0];
    qf[kc].y = p[1];
  }
  v8f o[8] = {};
  float mrow[8], lrow[8];
#pragma unroll
  for (int r = 0; r < 8; ++r) { mrow[r] = -__builtin_inff(); lrow[r] = 0.0f; }
  const float scale = 0.08838834764831845f;   // 1/sqrt(128)

  for (int kb = 0; kb < NODES; kb += 16) {
    // S = Q @ K^T for this 16-key chunk
    v8f s = {};
#pragma unroll
    for (int kc = 0; kc < 32; ++kc) {
      v2f b;
      const float* p = Km + (size_t)(g * NODES + kb + lo) * DM + cbase + kc * 4 + 2 * hi;
      b.x = p[0];
      b.y = p[1];
      s = WMMA_F32(qf[kc], b, s);
    }
    // online softmax update (rows r+8*hi, cols = 16 lanes of this half)
    float fac[8];
#pragma unroll
    for (int r = 0; r < 8; ++r) {
      float sv = s[r] * scale;
      float mx = sv;
      for (int d = 1; d < 16; d <<= 1) mx = fmaxf(mx, __shfl_xor(mx, d, 32));
      const float mnew = fmaxf(mrow[r], mx);
      const float f = __expf(mrow[r] - mnew);
      const float pv = __expf(sv - mnew);
      float rs = pv;
      for (int d = 1; d < 16; d <<= 1) rs += __shfl_xor(rs, d, 32);
      lrow[r] = lrow[r] * f + rs;
      mrow[r] = mnew;
      fac[r] = f;
      s[r] = pv;
    }
#pragma unroll
    for (int t = 0; t < 8; ++t) {
      v8f tmp = o[t];
#pragma unroll
      for (int r = 0; r < 8; ++r) tmp[r] *= fac[r];
      o[t] = tmp;
    }
    // transpose P (C-layout -> A-layout) through LDS
    __syncthreads();
#pragma unroll
    for (int r = 0; r < 8; ++r) Pt[r + 8 * hi][lo] = s[r];
    __syncthreads();
    v2f pf[4];
#pragma unroll
    for (int kc = 0; kc < 4; ++kc) {
      pf[kc].x = Pt[lo][kc * 4 + 2 * hi];
      pf[kc].y = Pt[lo][kc * 4 + 2 * hi + 1];
    }
    // O += P @ V_chunk
#pragma unroll
    for (int t = 0; t < 8; ++t) {
#pragma unroll
      for (int kc = 0; kc < 4; ++kc) {
        v2f b;
        const float* p = Vm + (size_t)(g * NODES + kb + kc * 4 + 2 * hi) * DM +
                         cbase + t * 16 + lo;
        b.x = p[0];
        b.y = p[DM];
        o[t] = WMMA_F32(pf[kc], b, o[t]);
      }
    }
  }
#pragma unroll
  for (int t = 0; t < 8; ++t)
#pragma unroll
    for (int r = 0; r < 8; ++r)
      O[(size_t)(rowbase + r + 8 * hi) * DM + cbase + t * 16 + lo] = o[t][r] / lrow[r];
}

// ---------------------------------------------------------------------------
// Fused FFN: out = relu(X @ W1 + b1) @ W2 + b2.  16-row tile / block (4 waves),
// hidden (16x1024) staged transposed in exactly 64KB LDS.
// ---------------------------------------------------------------------------
__global__ void ffn_fused(const float* __restrict__ X,
                          const float* __restrict__ W1, const float* __restrict__ b1,
                          const float* __restrict__ W2, const float* __restrict__ b2,
                          float* __restrict__ Out) {
  __shared__ float hid[FF][16];          // hid[col][row] = hidden[row][col]
  const int wave = threadIdx.x >> 5;
  const int lane = threadIdx.x & 31;
  const int lo = lane & 15, hi = lane >> 4;
  const int m0 = blockIdx.x * 16;

  // phase 1: hidden = relu(X @ W1 + b1)
  for (int t = wave; t < FF / 16; t += 4) {
    v8f acc = {};
    for (int k = 0; k < DM; k += 4) {
      v2f a;
      const float* ap = X + (size_t)(m0 + lo) * DM + k + 2 * hi;
      a.x = ap[0];
      a.y = ap[1];
      v2f b;
      const int col = t * 16 + lo;
      b.x = W1[(size_t)(k + 2 * hi) * FF + col];
      b.y = W1[(size_t)(k + 2 * hi + 1) * FF + col];
      acc = WMMA_F32(a, b, acc);
    }
    const int col = t * 16 + lo;
    const float bb = b1[col];
#pragma unroll
    for (int r = 0; r < 8; ++r) {
      const float v = acc[r] + bb;
      hid[col][r + 8 * hi] = v > 0.0f ? v : 0.0f;
    }
  }
  __syncthreads();
  // phase 2: out = hidden @ W2 + b2
  for (int t = wave; t < DM / 16; t += 4) {
    v8f acc = {};
    for (int k = 0; k < FF; k += 4) {
      v2f a;
      a.x = hid[k + 2 * hi][lo];
      a.y = hid[k + 2 * hi + 1][lo];
      v2f b;
      const int col = t * 16 + lo;
      b.x = W2[(size_t)(k + 2 * hi) * DM + col];
      b.y = W2[(size_t)(k + 2 * hi + 1) * DM + col];
      acc = WMMA_F32(a, b, acc);
    }
    const int col = t * 16 + lo;
    const float bb = b2[col];
#pragma unroll
    for (int r = 0; r < 8; ++r)
      Out[(size_t)(m0 + r + 8 * hi) * DM + col] = acc[r] + bb;
  }
}

// ---------------------------------------------------------------------------
// Residual + LayerNorm over DM=256: out = LN(X + Y) * g + b.  block = 256 = row.
// ---------------------------------------------------------------------------
__global__ void residual_ln(const float* __restrict__ X, const float* __restrict__ Y,
                            const float* __restrict__ g, const float* __restrict__ b,
                            float* __restrict__ out) {
  const int row = blockIdx.x;
  const int t = threadIdx.x;
  __shared__ float red[8];
  const float v = X[(size_t)row * DM + t] + Y[(size_t)row * DM + t];
  float s = v;
  for (int d = 1; d < 32; d <<= 1) s += __shfl_xor(s, d, 32);
  if ((t & 31) == 0) red[t >> 5] = s;
  __syncthreads();
  float mu = 0.0f;
#pragma unroll
  for (int i = 0; i < 8; ++i) mu += red[i];
  mu *= (1.0f / DM);
  __syncthreads();
  const float dv = v - mu;
  float sq = dv * dv;
  for (int d = 1; d < 32; d <<= 1) sq += __shfl_xor(sq, d, 32);
  if ((t & 31) == 0) red[t >> 5] = sq;
  __syncthreads();
  float var = 0.0f;
#pragma unroll
  for (int i = 0; i < 8; ++i) var += red[i];
  var *= (1.0f / DM);
  out[(size_t)row * DM + t] = dv * rsqrtf(var + 1e-5f) * g[t] + b[t];
}

// ---------------------------------------------------------------------------
// Mean pool per graph (exactly 1024 nodes per graph)
// ---------------------------------------------------------------------------
__global__ void pool_mean(const float* __restrict__ X, float* __restrict__ P) {
  const int idx = blockIdx.x * blockDim.x + threadIdx.x;   // g*256 + c
  if (idx >= NGRAPH * DM) return;
  const int g = idx >> 8, c = idx & 255;
  float s = 0.0f;
  const float* p = X + (size_t)g * NODES * DM + c;
  for (int r = 0; r < NODES; ++r) s += p[(size_t)r * DM];
  P[idx] = s * (1.0f / NODES);
}

// ---------------------------------------------------------------------------
// Host launcher
// ---------------------------------------------------------------------------
extern "C" void kernel_launch(void* const* d_in, const int* in_sizes, int n_in,
                              void* d_out, int out_size, void* d_ws, size_t ws_size,
                              hipStream_t stream) {
  const float* x     = (const float*)d_in[0];
  const int*   esrc  = (const int*)d_in[1];
  const int*   edst  = (const int*)d_in[2];
  const int    E     = in_sizes[1];
  const float* fc1_W = (const float*)d_in[4];
  const float* fc1_b = (const float*)d_in[5];
  const float* g0_W  = (const float*)d_in[6];
  const float* g0_as = (const float*)d_in[7];
  const float* g0_ad = (const float*)d_in[8];
  const float* g0_b  = (const float*)d_in[9];
  const float* gW    = (const float*)d_in[10];
  const float* gas   = (const float*)d_in[11];
  const float* gad   = (const float*)d_in[12];
  const float* gb    = (const float*)d_in[13];
  const float* sa_Wq = (const float*)d_in[14];
  const float* sa_bq = (const float*)d_in[15];
  const float* sa_Wk = (const float*)d_in[16];
  const float* sa_bk = (const float*)d_in[17];
  const float* sa_Wv = (const float*)d_in[18];
  const float* sa_bv = (const float*)d_in[19];
  const float* sa_Wo = (const float*)d_in[20];
  const float* sa_bo = (const float*)d_in[21];
  const float* ca_Wq = (const float*)d_in[22];
  const float* ca_bq = (const float*)d_in[23];
  const float* ca_Wk = (const float*)d_in[24];
  const float* ca_bk = (const float*)d_in[25];
  const float* ca_Wv = (const float*)d_in[26];
  const float* ca_bv = (const float*)d_in[27];
  const float* ca_Wo = (const float*)d_in[28];
  const float* ca_bo = (const float*)d_in[29];
  const float* ln1_g = (const float*)d_in[30];
  const float* ln1_b = (const float*)d_in[31];
  const float* ln2_g = (const float*)d_in[32];
  const float* ln2_b = (const float*)d_in[33];
  const float* ln3_g = (const float*)d_in[34];
  const float* ln3_b = (const float*)d_in[35];
  const float* ff1_W = (const float*)d_in[36];
  const float* ff1_b = (const float*)d_in[37];
  const float* ff2_W = (const float*)d_in[38];
  const float* ff2_b = (const float*)d_in[39];
  const float* fc_W  = (const float*)d_in[40];
  const float* fc_b  = (const float*)d_in[41];

  float* ws = (float*)d_ws;
  const size_t NB = (size_t)NN * DM;
  float* A  = ws;
  float* B  = ws + NB;
  float* C  = ws + 2 * NB;
  float* Qb = ws + 3 * NB;
  float* Kb = ws + 4 * NB;
  float* Vb = ws + 5 * NB;
  // edge scratch lives inside C (dead during the GAT phase)
  float*    ss    = C;
  float*    sd    = C + NN * HH;
  float*    eb    = C + 2 * NN * HH;
  unsigned* mkey  = (unsigned*)(C + 2 * NN * HH + (size_t)E * HH);
  float*    denom = C + 2 * NN * HH + (size_t)E * HH + NN * HH;

  const dim3 W32(32);

  // fc1: [16384,16] @ [16,64] + b  -> A (ld 64)
  gemm_wmma_f32<<<dim3(NN / 16, 1), W32, 0, stream>>>(x, fc1_W, fc1_b, A,
                                                      NN, 64, 16, 0);
  // ---- 4 GAT layers ----
  float* cur = A;
  float* nxt = B;
  for (int layer = 0; layer < 4; ++layer) {
    const float* Wl  = layer == 0 ? g0_W  : gW  + (size_t)(layer - 1) * DM * DM;
    const float* asl = layer == 0 ? g0_as : gas + (size_t)(layer - 1) * HH * CC;
    const float* adl = layer == 0 ? g0_ad : gad + (size_t)(layer - 1) * HH * CC;
    const float* bl  = layer == 0 ? g0_b  : gb  + (size_t)(layer - 1) * DM;
    const int K = layer == 0 ? 64 : DM;

    gemm_wmma_f32<<<dim3(NN / 16, DM / 64), W32, 0, stream>>>(cur, Wl, nullptr, Qb,
                                                              NN, DM, K, 0);
    gat_scores<<<(NN * HH + 255) / 256, 256, 0, stream>>>(Qb, asl, adl, ss, sd);
    hipMemsetAsync(mkey, 0, (size_t)NN * HH * 4, stream);
    hipMemsetAsync(denom, 0, (size_t)NN * HH * 4, stream);
    hipMemsetAsync(nxt, 0, NB * 4, stream);
    gat_edge1<<<(E * HH + 255) / 256, 256, 0, stream>>>(esrc, edst, ss, sd, eb, mkey, E);
    gat_edge2<<<(E * HH + 255) / 256, 256, 0, stream>>>(edst, eb, mkey, denom, E);
    gat_edge3<<<(unsigned)(((long long)E * DM + 255) / 256), 256, 0, stream>>>(
        esrc, edst, eb, denom, Qb, nxt, E);
    bias_leaky<<<(NN * DM + 255) / 256, 256, 0, stream>>>(nxt, bl, NN * DM);
    float* t = cur; cur = nxt; nxt = t;
  }
  float* T0 = cur;                       // transformer input / memory (== A)
  float* X1 = (cur == A) ? B : A;

  // ---- self-attention ----
  gemm_wmma_f32<<<dim3(NN / 16, DM / 64), W32, 0, stream>>>(T0, sa_Wq, sa_bq, Qb, NN, DM, DM, 0);
  gemm_wmma_f32<<<dim3(NN / 16, DM / 64), W32, 0, stream>>>(T0, sa_Wk, sa_bk, Kb, NN, DM, DM, 0);
  gemm_wmma_f32<<<dim3(NN / 16, DM / 64), W32, 0, stream>>>(T0, sa_Wv, sa_bv, Vb, NN, DM, DM, 0);
  flash_attn<<<dim3(NODES / 16, NHEAD, NGRAPH), W32, 0, stream>>>(Qb, Kb, Vb, C);
  gemm_wmma_f32<<<dim3(NN / 16, DM / 64), W32, 0, stream>>>(C, sa_Wo, sa_bo, X1, NN, DM, DM, 0);
  residual_ln<<<NN, DM, 0, stream>>>(T0, X1, ln1_g, ln1_b, C);       // C = t1

  // ---- cross-attention (q from t1, k/v from mem=T0) ----
  gemm_wmma_f32<<<dim3(NN / 16, DM / 64), W32, 0, stream>>>(C,  ca_Wq, ca_bq, Qb, NN, DM, DM, 0);
  gemm_wmma_f32<<<dim3(NN / 16, DM / 64), W32, 0, stream>>>(T0, ca_Wk, ca_bk, Kb, NN, DM, DM, 0);
  gemm_wmma_f32<<<dim3(NN / 16, DM / 64), W32, 0, stream>>>(T0, ca_Wv, ca_bv, Vb, NN, DM, DM, 0);
  flash_attn<<<dim3(NODES / 16, NHEAD, NGRAPH), W32, 0, stream>>>(Qb, Kb, Vb, X1);
  gemm_wmma_f32<<<dim3(NN / 16, DM / 64), W32, 0, stream>>>(X1, ca_Wo, ca_bo, T0, NN, DM, DM, 0);
  residual_ln<<<NN, DM, 0, stream>>>(C, T0, ln2_g, ln2_b, X1);       // X1 = t2

  // ---- FFN + LN3 ----
  ffn_fused<<<NN / 16, 128, 0, stream>>>(X1, ff1_W, ff1_b, ff2_W, ff2_b, C);
  residual_ln<<<NN, DM, 0, stream>>>(X1, C, ln3_g, ln3_b, T0);       // T0 = t3

  // ---- pool + final fc ----
  pool_mean<<<(NGRAPH * DM + 255) / 256, 256, 0, stream>>>(T0, Qb);  // pooled in Qb
  gemm_wmma_f32<<<dim3(1, OUTD / 64), W32, 0, stream>>>(Qb, fc_W, fc_b, (float*)d_out,
                                                        16, OUTD, DM, 0);
}